// LocalEnergyOpt_32770600469174
// MI455X (gfx1250) — compile-verified
//
#include <hip/hip_runtime.h>
#include <stdint.h>

// Problem constants (fixed by the reference).
#define MAXLEN  10000
#define NATOMS  2000
#define NBONDS  1999
#define NANG    1998
#define NTOR    1997
#define PROWS   350          // NBT + NAT + NTT = 50 + 100 + 200
#define FEPS    1e-8f

typedef __attribute__((ext_vector_type(2))) float v2f;
typedef __attribute__((ext_vector_type(8))) float v8f;

__global__ __launch_bounds__(256)
void energy_kernel(const float* __restrict__ features,
                   const int*   __restrict__ lengths,
                   const float* __restrict__ opt,
                   const float* __restrict__ btype,
                   const float* __restrict__ atype,
                   const float* __restrict__ ttype,
                   float* __restrict__ out)
{
    __shared__ float coord_s[3 * NATOMS];   // 24000 B
    __shared__ float opt_s[3 * PROWS];      //  4200 B
    __shared__ float red_s[256];
    __shared__ float red2[2];

    const int tid = threadIdx.x;
    const int b   = blockIdx.x;
    const float* fb = features + (size_t)b * MAXLEN * 9;

    // ---- CDNA5 async DMA: stage coords (col 5) and param table into LDS ----
    // GVS form: global_load_async_to_lds_b32 vDstLds, vOffset32, s[base64]
    unsigned cbase = (unsigned)(uintptr_t)(void*)coord_s;
    for (int m = tid; m < 3 * NATOMS; m += 256) {
        unsigned goff  = (unsigned)((m * 9 + 5) * 4);   // byte offset of feats[b,m,5]
        unsigned laddr = cbase + (unsigned)(m * 4);
        asm volatile("global_load_async_to_lds_b32 %0, %1, %2"
                     :: "v"(laddr), "v"(goff), "s"(fb) : "memory");
    }
    unsigned obase = (unsigned)(uintptr_t)(void*)opt_s;
    for (int m = tid; m < 3 * PROWS; m += 256) {
        asm volatile("global_load_async_to_lds_b32 %0, %1, %2"
                     :: "v"(obase + (unsigned)(m * 4)), "v"((unsigned)(m * 4)), "s"(opt)
                     : "memory");
    }
    asm volatile("s_wait_asynccnt 0" ::: "memory");
    __syncthreads();

    int nb = lengths[b * 9 + 6] / 3; if (nb > NBONDS) nb = NBONDS;
    int na = lengths[b * 9 + 7] / 4; if (na > NANG)   na = NANG;
    int nt = lengths[b * 9 + 8] / 5; if (nt > NTOR)   nt = NTOR;

    float esum = 0.0f;

    // ---- bonds: E = k * (r - r0)^2 ----
    for (int t = tid; t < nb; t += 256) {
        int i  = (int)fb[(3 * t + 0) * 9 + 6];
        int j  = (int)fb[(3 * t + 1) * 9 + 6];
        int ty = (int)fb[(3 * t + 2) * 9 + 6];
        float dx = coord_s[3 * i + 0] - coord_s[3 * j + 0];
        float dy = coord_s[3 * i + 1] - coord_s[3 * j + 1];
        float dz = coord_s[3 * i + 2] - coord_s[3 * j + 2];
        float r  = sqrtf(dx * dx + dy * dy + dz * dz + FEPS);
        int row  = (int)btype[ty];
        float kk = opt_s[3 * row + 0], r0 = opt_s[3 * row + 1];
        float d  = r - r0;
        esum += kk * d * d;
    }

    // ---- angles: E = k * (theta - theta0)^2 ----
    for (int t = tid; t < na; t += 256) {
        int i  = (int)fb[(4 * t + 0) * 9 + 7];
        int j  = (int)fb[(4 * t + 1) * 9 + 7];
        int k  = (int)fb[(4 * t + 2) * 9 + 7];
        int ty = (int)fb[(4 * t + 3) * 9 + 7];
        float ux = coord_s[3 * i + 0] - coord_s[3 * j + 0];
        float uy = coord_s[3 * i + 1] - coord_s[3 * j + 1];
        float uz = coord_s[3 * i + 2] - coord_s[3 * j + 2];
        float vx = coord_s[3 * k + 0] - coord_s[3 * j + 0];
        float vy = coord_s[3 * k + 1] - coord_s[3 * j + 1];
        float vz = coord_s[3 * k + 2] - coord_s[3 * j + 2];
        float uu = ux * ux + uy * uy + uz * uz + FEPS;
        float vv = vx * vx + vy * vy + vz * vz + FEPS;
        float uv = ux * vx + uy * vy + uz * vz;
        float c  = uv * rsqrtf(uu * vv);
        c = fminf(fmaxf(c, -1.0f + 1e-6f), 1.0f - 1e-6f);
        float th = acosf(c);
        int row  = (int)atype[ty];
        float kk = opt_s[3 * row + 0], t0 = opt_s[3 * row + 1];
        float d  = th - t0;
        esum += kk * d * d;
    }

    // ---- torsions: E = k * (1 + cos(n*phi - phi0)) ----
    for (int t = tid; t < nt; t += 256) {
        int i  = (int)fb[(5 * t + 0) * 9 + 8];
        int j  = (int)fb[(5 * t + 1) * 9 + 8];
        int k  = (int)fb[(5 * t + 2) * 9 + 8];
        int l  = (int)fb[(5 * t + 3) * 9 + 8];
        int ty = (int)fb[(5 * t + 4) * 9 + 8];
        float b1x = coord_s[3 * j + 0] - coord_s[3 * i + 0];
        float b1y = coord_s[3 * j + 1] - coord_s[3 * i + 1];
        float b1z = coord_s[3 * j + 2] - coord_s[3 * i + 2];
        float b2x = coord_s[3 * k + 0] - coord_s[3 * j + 0];
        float b2y = coord_s[3 * k + 1] - coord_s[3 * j + 1];
        float b2z = coord_s[3 * k + 2] - coord_s[3 * j + 2];
        float b3x = coord_s[3 * l + 0] - coord_s[3 * k + 0];
        float b3y = coord_s[3 * l + 1] - coord_s[3 * k + 1];
        float b3z = coord_s[3 * l + 2] - coord_s[3 * k + 2];
        float n1x = b1y * b2z - b1z * b2y;
        float n1y = b1z * b2x - b1x * b2z;
        float n1z = b1x * b2y - b1y * b2x;
        float n2x = b2y * b3z - b2z * b3y;
        float n2y = b2z * b3x - b2x * b3z;
        float n2z = b2x * b3y - b2y * b3x;
        float inv = rsqrtf(b2x * b2x + b2y * b2y + b2z * b2z + FEPS);
        float hx = b2x * inv, hy = b2y * inv, hz = b2z * inv;
        float m1x = n1y * hz - n1z * hy;
        float m1y = n1z * hx - n1x * hz;
        float m1z = n1x * hy - n1y * hx;
        float sy  = m1x * n2x + m1y * n2y + m1z * n2z;
        float sx  = n1x * n2x + n1y * n2y + n1z * n2z;
        float phi = atan2f(sy, sx);
        int row   = (int)ttype[ty];
        float kk  = opt_s[3 * row + 0];
        float p0  = opt_s[3 * row + 1];
        float nn  = opt_s[3 * row + 2];
        esum += kk * (1.0f + cosf(nn * phi - p0));
    }

    // ---- block reduction of 256 partials via V_WMMA_F32_16X16X4_F32 ----
    red_s[tid] = esum;
    __syncthreads();
    {
        // All 8 waves run the WMMA with full EXEC (required); results identical.
        const int lane = tid & 31;
        const int m    = lane & 15;
        const int kb   = (lane < 16) ? 0 : 2;   // A layout: VGPR0/1 = K {kb, kb+1}
        v2f ones; ones.x = 1.0f; ones.y = 1.0f; // B = 4x16 all-ones
        v8f acc = {};
#pragma unroll
        for (int c = 0; c < 4; ++c) {
            v2f a;
            a.x = red_s[c * 64 + m * 4 + kb + 0];
            a.y = red_s[c * 64 + m * 4 + kb + 1];
            // D[m][n] = sum_k A[m][k] + C[m][n]  -> row sums, accumulated over chunks
            acc = __builtin_amdgcn_wmma_f32_16x16x4_f32(
                false, a, false, ones, (short)0, acc, false, false);
        }
        float s = acc[0] + acc[1] + acc[2] + acc[3] + acc[4] + acc[5] + acc[6] + acc[7];
        if (tid == 0)  red2[0] = s;   // sum over M=0..7
        if (tid == 16) red2[1] = s;   // sum over M=8..15
    }
    __syncthreads();
    if (tid == 0) out[b] = red2[0] + red2[1];
}

extern "C" void kernel_launch(void* const* d_in, const int* in_sizes, int n_in,
                              void* d_out, int out_size, void* d_ws, size_t ws_size,
                              hipStream_t stream) {
    const float* features = (const float*)d_in[0];
    const int*   lengths  = (const int*)d_in[1];
    const float* opt      = (const float*)d_in[2];
    const float* btype    = (const float*)d_in[3];
    const float* atype    = (const float*)d_in[4];
    const float* ttype    = (const float*)d_in[5];
    float* out = (float*)d_out;

    const int Bn = in_sizes[0] / (MAXLEN * 9);   // = 256
    energy_kernel<<<dim3(Bn), dim3(256), 0, stream>>>(
        features, lengths, opt, btype, atype, ttype, out);
}